// AttentionPointSelector_1752346656929
// MI455X (gfx1250) — compile-verified
//
#include <hip/hip_runtime.h>
#include <hip/hip_bf16.h>

typedef __attribute__((ext_vector_type(2))) float v2f;
typedef __attribute__((ext_vector_type(8))) float v8f;
typedef __attribute__((ext_vector_type(4))) int   v4i;

// addrspace-qualified int4 pointers for the async global->LDS builtin
typedef __attribute__((address_space(1))) v4i gas_v4i;   // global (__device__)
typedef __attribute__((address_space(3))) v4i las_v4i;   // LDS (__shared__)

#define B_    2
#define C_    64
#define T_    16
#define PN_   512
#define D_    1024      // T_*C_
#define THW_  65536     // 16*64*64
#define TOPK_ 128

#define KC_      32     // K-chunk staged in LDS per iteration
#define NCHUNK_  (D_ / KC_)
#define PSTRIDE_ 36     // LDS panel row stride (floats): 144B -> float4-aligned,
                        // and 36*r mod 64 distinct for r=0..15 -> no bank conflicts
#define PANEL_F_ (PN_ * PSTRIDE_)                  // 18,432 floats = 73,728 B
#define SIM_F_   (16 * PN_)                        //  8,192 floats = 32,768 B
#define SMEM_B_  ((2 * PANEL_F_ + SIM_F_) * 4)     // 180,224 B dynamic LDS (<320KB/WGP)

#if defined(__HIP_DEVICE_COMPILE__) && __has_builtin(__builtin_amdgcn_global_load_async_to_lds_b128)
#define HAVE_ASYNC_LDS 1
#else
#define HAVE_ASYNC_LDS 0
#endif

#if defined(__HIP_DEVICE_COMPILE__) && __has_builtin(__builtin_amdgcn_s_wait_asynccnt)
#define WAIT_ASYNC0() __builtin_amdgcn_s_wait_asynccnt(0)
#else
#define WAIT_ASYNC0() asm volatile("s_wait_asynccnt 0x0" ::: "memory")
#endif

// ---------------------------------------------------------------------------
// K1: x[b,c,t,pn] -> xr[b,pn, t*C+c]   (coalesced writes; 4 MB total)
// ---------------------------------------------------------------------------
__global__ __launch_bounds__(256) void k_transpose(const float* __restrict__ x,
                                                   float* __restrict__ xr) {
  int o   = blockIdx.x * 256 + threadIdx.x;   // B_*PN_*D_ = 2^20 exactly
  int b   = o >> 19;
  int rem = o & 524287;
  int pn  = rem >> 10;
  int d   = rem & 1023;
  int t   = d >> 6;
  int c   = d & 63;
  xr[o] = x[(((b * C_ + c) * T_ + t) << 9) + pn];
}

// ---------------------------------------------------------------------------
// K2: per (batch, 16-row stripe): sim = xr.xr^T/32 via fp32 WMMA fed from a
//     double-buffered LDS K-panel filled with async global->LDS loads, then
//     fused softmax + row-mean (sim never touches HBM).
//     8 waves/WG; wave w owns j-tiles w*4..w*4+3 (4 x v8f accumulators).
// ---------------------------------------------------------------------------
__global__ __launch_bounds__(256) void k_gram_softmax(const float* __restrict__ xr,
                                                      float* __restrict__ scores) {
  extern __shared__ float smem[];
  float* panel0 = smem;                    // [PN_][PSTRIDE_] buffer A
  float* panel1 = smem + PANEL_F_;         // [PN_][PSTRIDE_] buffer B
  float* sim    = smem + 2 * PANEL_F_;     // [16][PN_] result stripe

  const int b    = blockIdx.x >> 5;           // 32 row-blocks per batch
  const int i0   = (blockIdx.x & 31) << 4;
  const int tid  = threadIdx.x;
  const int w    = tid >> 5;                  // wave id 0..7
  const int l    = tid & 31;                  // lane id
  const int r    = l & 15;                    // row-in-tile this lane loads
  const int koff = (l >> 4) << 1;             // fp32 16x4 A/B layout: K=0,1 | 2,3

  // per-lane LDS fragment offsets (floats, relative to a panel buffer)
  const int aoff = (i0 + r) * PSTRIDE_ + koff;
  int boff[4];
#pragma unroll
  for (int q = 0; q < 4; ++q)
    boff[q] = ((((w << 2) + q) << 4) + r) * PSTRIDE_ + koff;

  // cooperative loader mapping: 8 threads x float4 cover one 32-float row
  const int lf = tid & 7;                     // float4 slot within row
  const int lr = tid >> 3;                    // row base (0..31), +32 per pass

  v8f zero = {};
  v8f acc[4] = {zero, zero, zero, zero};

  // ---- WMMA compute on one staged panel buffer ----
  // Register-pipelined: B-fragments of step s+1 are loaded before the WMMAs of
  // step s issue, keeping >=4 ds_load_b64 in flight (partial s_wait_dscnt).
  auto do_compute = [&](const float* buf) {
    v2f a[8];
#pragma unroll
    for (int s = 0; s < 8; ++s)
      a[s] = *(const v2f*)(buf + aoff + (s << 2));

    v2f cur[4], nxt[4];
#pragma unroll
    for (int q = 0; q < 4; ++q)
      cur[q] = *(const v2f*)(buf + boff[q]);

#pragma unroll
    for (int s = 0; s < 8; ++s) {
      if (s < 7) {
#pragma unroll
        for (int q = 0; q < 4; ++q)
          nxt[q] = *(const v2f*)(buf + boff[q] + ((s + 1) << 2));
      }
#pragma unroll
      for (int q = 0; q < 4; ++q)
        acc[q] = __builtin_amdgcn_wmma_f32_16x16x4_f32(
            /*neg_a=*/false, a[s], /*neg_b=*/false, cur[q],
            /*c_mod=*/(short)0, acc[q], /*reuse_a=*/false, /*reuse_b=*/false);
      if (s < 7) {
#pragma unroll
        for (int q = 0; q < 4; ++q) cur[q] = nxt[q];  // SSA copies, elided
      }
    }
  };

#if HAVE_ASYNC_LDS
  // ---- async global->LDS staging (ASYNCcnt), double-buffered ----
  auto issue_chunk = [&](int kc, float* buf) {
#pragma unroll
    for (int p = 0; p < 16; ++p) {
      const int row = lr + (p << 5);
      const float* g = xr + ((size_t)(b * PN_ + row)) * D_ + kc + (lf << 2);
      float* s = buf + row * PSTRIDE_ + (lf << 2);
      __builtin_amdgcn_global_load_async_to_lds_b128(
          (gas_v4i*)(unsigned long long)g,
          (las_v4i*)(unsigned int)(unsigned long long)s,
          0, 0);
    }
  };

  issue_chunk(0, panel0);
  for (int c = 0; c < NCHUNK_; ++c) {
    float* bufc = (c & 1) ? panel1 : panel0;
    float* bufn = (c & 1) ? panel0 : panel1;
    WAIT_ASYNC0();        // my async loads into bufc have landed in LDS
    __syncthreads();      // everyone's loads landed; bufn free (prev compute done)
    if (c + 1 < NCHUNK_) issue_chunk((c + 1) * KC_, bufn);  // overlaps WMMAs below
    do_compute(bufc);
  }
  __syncthreads();        // protect sim region writes below vs. last reads
#else
  // ---- fallback: synchronous staging through VGPRs ----
  for (int kc = 0; kc < D_; kc += KC_) {
#pragma unroll 4
    for (int p = 0; p < 16; ++p) {
      const int row = lr + (p << 5);
      const float4 vv =
          ((const float4*)(xr + ((size_t)(b * PN_ + row)) * D_ + kc))[lf];
      *(float4*)(panel0 + row * PSTRIDE_ + (lf << 2)) = vv;
    }
    __syncthreads();
    do_compute(panel0);
    __syncthreads();
  }
#endif

  // Scatter C tiles (VGPR v: lanes 0-15 -> M=v, lanes 16-31 -> M=v+8) into LDS
  const int mbase = (l >> 4) << 3;
  const int n     = l & 15;
#pragma unroll
  for (int q = 0; q < 4; ++q) {
    const int j0 = (((w << 2) + q) << 4);
#pragma unroll
    for (int v = 0; v < 8; ++v)
      sim[(mbase + v) * PN_ + j0 + n] = acc[q][v] * 0.03125f;  // * d_k^-0.5
  }
  __syncthreads();

  // Fused softmax + mean: wave w handles rows w and w+8 of the stripe.
  for (int row = w; row < 16; row += 8) {
    float* sr = sim + row * PN_;
    float m = -3.4e38f;
    for (int c = l; c < PN_; c += 32) m = fmaxf(m, sr[c]);
#pragma unroll
    for (int off = 16; off > 0; off >>= 1) m = fmaxf(m, __shfl_xor(m, off, 32));

    float s = 0.f;
    for (int c = l; c < PN_; c += 32) {
      float e = __expf(sr[c] - m);
      sr[c] = e;
      s += e;
    }
#pragma unroll
    for (int off = 16; off > 0; off >>= 1) s += __shfl_xor(s, off, 32);

    float s2 = 0.f;  // mean of attn row = sum(e/s) / PN (element-wise div, like jnp)
    for (int c = l; c < PN_; c += 32) s2 += sr[c] / s;
#pragma unroll
    for (int off = 16; off > 0; off >>= 1) s2 += __shfl_xor(s2, off, 32);

    if (l == 0) scores[b * PN_ + i0 + row] = s2 * (1.0f / (float)PN_);
  }
}

// ---------------------------------------------------------------------------
// K3: exact top-128 per batch via ranking (matches lax.top_k tie semantics:
//     descending values, ties broken by ascending index).
// ---------------------------------------------------------------------------
__global__ __launch_bounds__(512) void k_topk(const float* __restrict__ scores,
                                              int* __restrict__ topk) {
  __shared__ float sc[PN_];
  const int b = blockIdx.x;
  const int t = threadIdx.x;
  sc[t] = scores[b * PN_ + t];
  __syncthreads();
  const float v = sc[t];
  int rank = 0;
  for (int j = 0; j < PN_; ++j) {
    float u = sc[j];
    rank += (u > v) || (u == v && j < t);
  }
  if (rank < TOPK_) topk[b * TOPK_ + rank] = t;
}

// ---------------------------------------------------------------------------
// K4: gather selected trajectory maps with b128 streaming copies.
//     One block per (b,k): 256 KB copy, float4 per lane, 4 loads in flight.
// ---------------------------------------------------------------------------
__global__ __launch_bounds__(256) void k_gather(const float* __restrict__ traj,
                                                const int* __restrict__ topk,
                                                float* __restrict__ out) {
  const int bk = blockIdx.x;            // b*TOPK_ + k
  const int b  = bk >> 7;
  const int pn = topk[bk];              // uniform per block -> scalar load
  const float4* __restrict__ src =
      (const float4*)(traj + ((size_t)(b * PN_) + pn) * THW_);
  float4* __restrict__ dst = (float4*)(out + (size_t)bk * THW_);
#pragma unroll 4
  for (int i = threadIdx.x; i < THW_ / 4; i += 256) dst[i] = src[i];
}

// ---------------------------------------------------------------------------
extern "C" void kernel_launch(void* const* d_in, const int* in_sizes, int n_in,
                              void* d_out, int out_size, void* d_ws, size_t ws_size,
                              hipStream_t stream) {
  const float* x    = (const float*)d_in[0];   // [2,64,16,512]
  const float* traj = (const float*)d_in[1];   // [2,512,16,64,64]
  float* out        = (float*)d_out;           // [2,128,16,64,64]

  float* wsf    = (float*)d_ws;
  float* xr     = wsf;                                   // B*PN*D   = 1,048,576 f
  float* scores = wsf + (size_t)B_ * PN_ * D_;           // B*PN     = 1,024 f
  int*   topk   = (int*)(scores + B_ * PN_);             // B*TOPK   = 256 i

  k_transpose   <<<(B_ * PN_ * D_) / 256, 256, 0, stream>>>(x, xr);
  k_gram_softmax<<<B_ * (PN_ / 16),       256, SMEM_B_, stream>>>(xr, scores);
  k_topk        <<<B_,                    512, 0, stream>>>(scores, topk);
  k_gather      <<<B_ * TOPK_,            256, 0, stream>>>(traj, topk, out);
}